// MlpAttention_20229295964392
// MI455X (gfx1250) — compile-verified
//
#include <hip/hip_runtime.h>
#include <hip/hip_bf16.h>
#include <math.h>

// ---------------------------------------------------------------------------
// MLP (additive) attention for MI455X / gfx1250.
//
// Sizes: B=16, S=4096, Q=H=V=512, f32.
// Roofline: dominant traffic = projected_keys (134 MB) + values (134 MB)
// read exactly once  =>  ~11.5 us at 23.3 TB/s.  Total FLOPs ~0.2 GFLOP =>
// compute is negligible; keep everything f32 and optimize data movement.
// WMMA is used where it genuinely applies: the [16,512]x[512,512] query
// projection via V_WMMA_F32_16X16X4_F32 (full-precision f32 matrix op).
// ---------------------------------------------------------------------------

#define B_  16
#define S_  4096
#define H_  512
#define V_  512

typedef float v2f __attribute__((ext_vector_type(2)));
typedef float v8f __attribute__((ext_vector_type(8)));

__device__ __forceinline__ float dev_tanh(float x) {
#if __has_builtin(__builtin_amdgcn_tanhf)
  return __builtin_amdgcn_tanhf(x);          // gfx1250 V_TANH_F32 (trans op)
#else
  float ax = fabsf(x);
  float e  = __expf(-2.0f * ax);             // v_exp_f32 path
  float t  = (1.0f - e) / (1.0f + e);
  return copysignf(t, x);
#endif
}

// ---------------------------------------------------------------------------
// K1: projected_query[b,h] = sum_q query[b,q] * w_query[q,h]
// M=16 (=B) x N=512 x K=512 GEMM with V_WMMA_F32_16X16X4_F32.
// One wave (32 lanes) per 16-wide H tile; 128 WMMA steps over K.
// A layout (16x4 f32, ISA 7.12.2): lane l, m=l&15, kk=2*(l>>4):
//   VGPR0 = A[m][k+kk], VGPR1 = A[m][k+kk+1]
// B layout (4x16 f32, rows striped across lanes): lane l, n=l&15:
//   VGPR0 = B[k+kk][n], VGPR1 = B[k+kk+1][n]
// C/D (16x16 f32, 8 VGPRs): VGPR r, lane l -> M = r + 8*(l>>4), N = l&15.
// ---------------------------------------------------------------------------
__global__ void k_project_wmma(const float* __restrict__ query,
                               const float* __restrict__ wq,
                               float* __restrict__ pq) {
  const int lane = threadIdx.x;            // 0..31, EXEC all ones (no divergence)
  const int h0   = blockIdx.x * 16;        // H tile
  const int m    = lane & 15;
  const int n    = lane & 15;
  const int kk   = (lane >> 4) * 2;        // 0 or 2

  v8f c = {};
  for (int k = 0; k < H_; k += 4) {
    v2f a, b;
    const float* ap = query + m * H_ + (k + kk);   // contiguous pair, 8B aligned
    a.x = ap[0];
    a.y = ap[1];
    b.x = wq[(size_t)(k + kk)     * H_ + h0 + n];
    b.y = wq[(size_t)(k + kk + 1) * H_ + h0 + n];
    // 8 args: (neg_a, A, neg_b, B, c_mod, C, reuse_a, reuse_b)
    c = __builtin_amdgcn_wmma_f32_16x16x4_f32(false, a, false, b,
                                              (short)0, c, false, false);
  }

  const int mrow = 8 * (lane >> 4);
#pragma unroll
  for (int r = 0; r < 8; ++r)
    pq[(size_t)(mrow + r) * H_ + h0 + n] = c[r];
}

// ---------------------------------------------------------------------------
// K2: raw scores. One wave per (b,s) row (4 rows per wave to amortize the
// pq / w_energy register preload). Streams projected_keys once with b128
// loads: lane covers h = i*128 + lane*4 .. +3, i=0..3 (512 f32 per row).
// Wave32 reduction via __shfl_xor (ds_swizzle path).
// ---------------------------------------------------------------------------
__global__ void k_scores(const float* __restrict__ pk,
                         const float* __restrict__ pq,
                         const float* __restrict__ we,
                         const unsigned char* __restrict__ mask,
                         float* __restrict__ scores) {
  const int lane = threadIdx.x & 31;
  const int wave = threadIdx.x >> 5;
  const int gw   = blockIdx.x * 8 + wave;   // global wave id
  const int r0   = gw * 4;                  // first of 4 rows (same b: 4 | 4096)
  const int b    = r0 >> 12;                // r0 / S_

  float4 vwe[4], vpq[4];
#pragma unroll
  for (int i = 0; i < 4; ++i) {
    const int h = i * 128 + lane * 4;
    vwe[i] = *(const float4*)(we + h);
    vpq[i] = *(const float4*)(pq + (size_t)b * H_ + h);
  }

  for (int rr = 0; rr < 4; ++rr) {
    const int row = r0 + rr;
    const float* prow = pk + (size_t)row * H_;
    float acc = 0.0f;
#pragma unroll
    for (int i = 0; i < 4; ++i) {
      const int h = i * 128 + lane * 4;
      float4 v = *(const float4*)(prow + h);      // global_load_b128
      acc += dev_tanh(v.x + vpq[i].x) * vwe[i].x;
      acc += dev_tanh(v.y + vpq[i].y) * vwe[i].y;
      acc += dev_tanh(v.z + vpq[i].z) * vwe[i].z;
      acc += dev_tanh(v.w + vpq[i].w) * vwe[i].w;
    }
#pragma unroll
    for (int off = 16; off > 0; off >>= 1)
      acc += __shfl_xor(acc, off, 32);
    if (lane == 0)
      scores[row] = mask[row] ? acc : -INFINITY;
  }
}

// ---------------------------------------------------------------------------
// K3: in-place softmax over S per batch. 16 blocks x 1024 threads, 4 values
// per thread, shared-memory tree reductions.
// ---------------------------------------------------------------------------
__global__ void k_softmax(float* __restrict__ scores) {
  const int b = blockIdx.x;
  const int t = threadIdx.x;                 // 0..1023
  float* s = scores + (size_t)b * S_;

  float x[4];
  float m = -INFINITY;
#pragma unroll
  for (int i = 0; i < 4; ++i) {
    x[i] = s[t + i * 1024];
    m = fmaxf(m, x[i]);
  }

  __shared__ float red[1024];
  red[t] = m;
  __syncthreads();
  for (int off = 512; off > 0; off >>= 1) {
    if (t < off) red[t] = fmaxf(red[t], red[t + off]);
    __syncthreads();
  }
  const float smax = red[0];
  __syncthreads();

  float e[4];
  float lsum = 0.0f;
#pragma unroll
  for (int i = 0; i < 4; ++i) {
    e[i] = __expf(x[i] - smax);              // exp(-inf - max) -> 0 for masked
    lsum += e[i];
  }
  red[t] = lsum;
  __syncthreads();
  for (int off = 512; off > 0; off >>= 1) {
    if (t < off) red[t] += red[t + off];
    __syncthreads();
  }
  const float inv = 1.0f / red[0];
#pragma unroll
  for (int i = 0; i < 4; ++i)
    s[t + i * 1024] = e[i] * inv;
}

// ---------------------------------------------------------------------------
// K4: context partials. Block = (b, s-chunk of 64); 256 threads x float2
// cover all V=512 columns, so each iteration reads 2 KB of `values`
// fully coalesced. Probabilities for the chunk are staged in LDS.
// 1024 blocks give enough concurrency to saturate HBM. Deterministic
// (no atomics): partials go to workspace, reduced by K5.
// ---------------------------------------------------------------------------
__global__ void k_context_partial(const float* __restrict__ values,
                                  const float* __restrict__ p,
                                  float* __restrict__ partial) {
  const int chunk = blockIdx.x & 63;
  const int b     = blockIdx.x >> 6;
  const int t     = threadIdx.x;             // 0..255

  __shared__ float sp[64];
  if (t < 64) sp[t] = p[(size_t)b * S_ + chunk * 64 + t];
  __syncthreads();

  const float* vb = values + ((size_t)b * S_ + (size_t)chunk * 64) * V_;
  float2 acc = make_float2(0.0f, 0.0f);
#pragma unroll 4
  for (int i = 0; i < 64; ++i) {
    float2 v = *(const float2*)(vb + (size_t)i * V_ + t * 2);
    const float w = sp[i];
    acc.x += w * v.x;
    acc.y += w * v.y;
  }
  *(float2*)(partial + (size_t)blockIdx.x * V_ + t * 2) = acc;
}

// K5: context[b,v] = sum over 64 chunks of partial[(b*64+c)*512 + v]
__global__ void k_context_reduce(const float* __restrict__ partial,
                                 float* __restrict__ ctx) {
  const int idx = blockIdx.x * blockDim.x + threadIdx.x;  // 0..8191
  const int b = idx >> 9;
  const int v = idx & 511;
  float acc = 0.0f;
#pragma unroll 8
  for (int c = 0; c < 64; ++c)
    acc += partial[((size_t)(b * 64 + c)) * V_ + v];
  ctx[idx] = acc;
}

// ---------------------------------------------------------------------------
extern "C" void kernel_launch(void* const* d_in, const int* in_sizes, int n_in,
                              void* d_out, int out_size, void* d_ws, size_t ws_size,
                              hipStream_t stream) {
  const float*         query  = (const float*)d_in[0];          // [16,1,512]
  const float*         pk     = (const float*)d_in[1];          // [16,4096,512]
  const float*         values = (const float*)d_in[2];          // [16,4096,512]
  const unsigned char* mask   = (const unsigned char*)d_in[3];  // [16,4096] bool
  const float*         wq     = (const float*)d_in[4];          // [512,512]
  const float*         we     = (const float*)d_in[5];          // [512,1]

  float* out    = (float*)d_out;
  float* ctx    = out;                 // [16,512]   (output 0)
  float* scores = out + B_ * V_;       // [16,4096]  (output 1)

  // Workspace layout (floats): pq [16*512], partials [1024*512]  (~2.1 MB)
  float* pq      = (float*)d_ws;
  float* partial = pq + B_ * H_;

  k_project_wmma    <<<H_ / 16,  32,  0, stream>>>(query, wq, pq);
  k_scores          <<<2048,     256, 0, stream>>>(pk, pq, we, mask, scores);
  k_softmax         <<<B_,       1024,0, stream>>>(scores);
  k_context_partial <<<B_ * 64,  256, 0, stream>>>(values, scores, partial);
  k_context_reduce  <<<32,       256, 0, stream>>>(partial, ctx);
}